// DiagSheafs_2594160246965
// MI455X (gfx1250) — compile-verified
//
#include <hip/hip_runtime.h>
#include <hip/hip_bf16.h>

// ---------------------------------------------------------------------------
// Problem constants (match the reference)
// ---------------------------------------------------------------------------
#define NN   20000     // nodes
#define EE   2000      // hyperedges
#define NNZ  100000    // incidence pairs
#define FF   128       // features
#define HH   128       // hidden
#define DD   6         // sheaf dim
#define COUT 40        // classes
#define DH   (DD * HH) // 768

typedef __attribute__((ext_vector_type(2))) float v2f;
typedef __attribute__((ext_vector_type(8))) float v8f;

#define ATOMIC_ADD_F32(p, v) unsafeAtomicAdd((p), (v))

// ---------------------------------------------------------------------------
// WMMA fp32 GEMM:  C[M,Nc] = A[M,K] @ B[K,Nc]
//   - 128 threads (4 wave32); each block computes a 32(M) x 64(N) tile
//   - each wave owns a 32x16 column slab: two v8f accumulators sharing each
//     B fragment (halves B LDS traffic, 16 v_wmma per k-tile)
//   - K must be a multiple of 32 (128 / 768 at all call sites)
//   - M and Nc are bounds-guarded (handles Nc = 6, 40, 128, 768; M = 2000)
//   - prefetches the next A k-tile (global_prefetch_b8) while computing
// ---------------------------------------------------------------------------
#define TK 32
#define TM 32
__global__ __launch_bounds__(128)
void gemm_wmma_f32(const float* __restrict__ A, const float* __restrict__ B,
                   float* __restrict__ C, int M, int Nc, int K, int ldb, int ldc) {
  __shared__ float As[TM][TK + 1];   // +1 pad: conflict-free row reads
  __shared__ float Bs[TK][64];

  const int tid  = threadIdx.x;
  const int wave = tid >> 5;         // 0..3  (wave32)
  const int lane = tid & 31;
  const int la   = lane & 15;
  const int hi   = lane >> 4;        // 0: K pair {0,1}, 1: K pair {2,3}

  const int m0 = blockIdx.y * TM;
  const int nb = blockIdx.x * 64;
  const int n0 = nb + wave * 16;

  v8f acc0 = {};                     // rows m0 .. m0+15
  v8f acc1 = {};                     // rows m0+16 .. m0+31

  for (int kk = 0; kk < K; kk += TK) {
    // cooperative loads: A tile TMxTK (1024 f32), B tile TKx64 (2048 f32)
    for (int i = tid; i < TM * TK; i += 128) {
      int r = i / TK, c = i % TK;
      int gm = m0 + r;
      As[r][c] = (gm < M) ? A[(size_t)gm * K + kk + c] : 0.0f;
    }
    for (int i = tid; i < TK * 64; i += 128) {
      int r = i >> 6, c = i & 63;
      int gn = nb + c;
      Bs[r][c] = (gn < Nc) ? B[(size_t)(kk + r) * ldb + gn] : 0.0f;
    }
    // prefetch next A k-tile into near caches while this tile computes
    if (kk + TK < K) {
      int pr = tid >> 2, pc = (tid & 3) << 3;       // 32 rows x 4 segments
      int gm = m0 + pr;
      if (gm < M) __builtin_prefetch(&A[(size_t)gm * K + kk + TK + pc], 0, 3);
    }
    __syncthreads();

#pragma unroll
    for (int k4 = 0; k4 < TK; k4 += 4) {
      v2f a0, a1, b;
      b.x  = Bs[k4 + 2 * hi][(wave << 4) + la];
      b.y  = Bs[k4 + 2 * hi + 1][(wave << 4) + la];
      a0.x = As[la][k4 + 2 * hi];
      a0.y = As[la][k4 + 2 * hi + 1];
      a1.x = As[16 + la][k4 + 2 * hi];
      a1.y = As[16 + la][k4 + 2 * hi + 1];
      acc0 = __builtin_amdgcn_wmma_f32_16x16x4_f32(
          false, a0, false, b, (short)0, acc0, false, false);
      acc1 = __builtin_amdgcn_wmma_f32_16x16x4_f32(
          false, a1, false, b, (short)0, acc1, false, false);
    }
    __syncthreads();
  }

  // store D: vgpr r -> row (+ 8*hi), col n0 + la
  const int n = n0 + la;
  if (n < Nc) {
#pragma unroll
    for (int r = 0; r < 8; ++r) {
      int gm = m0 + r + 8 * hi;
      if (gm < M) C[(size_t)gm * ldc + n] = acc0[r];
    }
#pragma unroll
    for (int r = 0; r < 8; ++r) {
      int gm = m0 + 16 + r + 8 * hi;
      if (gm < M) C[(size_t)gm * ldc + n] = acc1[r];
    }
  }
}

// ---------------------------------------------------------------------------
// mean over the d axis of [rows, D, F] -> [rows, F]
// ---------------------------------------------------------------------------
__global__ void mean_d_kernel(const float* __restrict__ xl, float* __restrict__ xm,
                              int rows) {
  int gid = blockIdx.x * blockDim.x + threadIdx.x;
  if (gid >= rows * FF) return;
  int n = gid >> 7, f = gid & 127;
  const float* p = xl + (size_t)n * DH + f;
  float s = 0.0f;
#pragma unroll
  for (int d = 0; d < DD; ++d) s += p[d * FF];
  xm[gid] = s * (1.0f / (float)DD);
}

// ---------------------------------------------------------------------------
// alpha = sigmoid(xm[row] + em[col]); also accumulate node/edge sheaf degrees
// ---------------------------------------------------------------------------
__global__ void alpha_deg_kernel(const float* __restrict__ xm, const float* __restrict__ em,
                                 const int* __restrict__ row, const int* __restrict__ col,
                                 float* __restrict__ alpha, float* __restrict__ Dg,
                                 float* __restrict__ Bg) {
  int gid = blockIdx.x * blockDim.x + threadIdx.x;
  if (gid >= NNZ * DD) return;
  int i = gid / DD, dd = gid - i * DD;
  int r = row[i], c = col[i];
  float v = xm[r * DD + dd] + em[c * DD + dd];
  v = 1.0f / (1.0f + __expf(-v));
  alpha[gid] = v;
  ATOMIC_ADD_F32(&Dg[r * DD + dd], v);
  ATOMIC_ADD_F32(&Bg[c * DD + dd], v);
}

__global__ void recip_kernel(float* __restrict__ p, int n) {
  int gid = blockIdx.x * blockDim.x + threadIdx.x;
  if (gid >= n) return;
  float v = p[gid];
  p[gid] = (v != 0.0f) ? (1.0f / v) : 0.0f;
}

__global__ void fill_kernel(float* __restrict__ p, size_t n, float v) {
  size_t stride = (size_t)gridDim.x * blockDim.x;
  for (size_t i = blockIdx.x * (size_t)blockDim.x + threadIdx.x; i < n; i += stride)
    p[i] = v;
}

// ---------------------------------------------------------------------------
// node -> hyperedge:  m[col] += (Binv[col]*alpha)[dd] * xp[row]
// one block per incidence; 192 threads x float4 = 768 floats (d x H block)
// ---------------------------------------------------------------------------
__global__ __launch_bounds__(192)
void scatter_to_edge(const float* __restrict__ xp, const float* __restrict__ alpha,
                     const float* __restrict__ Binv, const int* __restrict__ row,
                     const int* __restrict__ col, float* __restrict__ m) {
  int i = blockIdx.x;
  int idx = threadIdx.x << 2;          // 0..764, stays within one dd (H=128 % 4 == 0)
  int dd = idx >> 7;
  int r = row[i], c = col[i];
  float s = Binv[c * DD + dd] * alpha[(size_t)i * DD + dd];
  float4 v = *(const float4*)(xp + (size_t)r * DH + idx);
  float* dst = m + (size_t)c * DH + idx;
  ATOMIC_ADD_F32(dst + 0, s * v.x);
  ATOMIC_ADD_F32(dst + 1, s * v.y);
  ATOMIC_ADD_F32(dst + 2, s * v.z);
  ATOMIC_ADD_F32(dst + 3, s * v.w);
}

// hyperedge -> node:  out[row] += (Dinv[row]*alpha)[dd] * m[col]
__global__ __launch_bounds__(192)
void scatter_to_node(const float* __restrict__ m, const float* __restrict__ alpha,
                     const float* __restrict__ Dinv, const int* __restrict__ row,
                     const int* __restrict__ col, float* __restrict__ out) {
  int i = blockIdx.x;
  int idx = threadIdx.x << 2;
  int dd = idx >> 7;
  int r = row[i], c = col[i];
  float s = Dinv[r * DD + dd] * alpha[(size_t)i * DD + dd];
  float4 v = *(const float4*)(m + (size_t)c * DH + idx);
  float* dst = out + (size_t)r * DH + idx;
  ATOMIC_ADD_F32(dst + 0, s * v.x);
  ATOMIC_ADD_F32(dst + 1, s * v.y);
  ATOMIC_ADD_F32(dst + 2, s * v.z);
  ATOMIC_ADD_F32(dst + 3, s * v.w);
}

// h = (elu?)(h + b[h_idx % H])
__global__ void bias_act_kernel(float* __restrict__ h, const float* __restrict__ b,
                                size_t total, int do_elu) {
  size_t gid = blockIdx.x * (size_t)blockDim.x + threadIdx.x;
  if (gid >= total) return;
  float v = h[gid] + b[gid & 127];
  if (do_elu) v = (v > 0.0f) ? v : (__expf(v) - 1.0f);
  h[gid] = v;
}

// ---------------------------------------------------------------------------
static inline void launch_gemm(const float* A, const float* B, float* C,
                               int M, int Nc, int K, int ldb, int ldc,
                               hipStream_t s) {
  dim3 grid((Nc + 63) / 64, (M + TM - 1) / TM);
  gemm_wmma_f32<<<grid, dim3(128), 0, s>>>(A, B, C, M, Nc, K, ldb, ldc);
}

extern "C" void kernel_launch(void* const* d_in, const int* in_sizes, int n_in,
                              void* d_out, int out_size, void* d_ws, size_t ws_size,
                              hipStream_t stream) {
  const float* x        = (const float*)d_in[0];
  const int*   eidx     = (const int*)d_in[1];
  const float* hedge    = (const float*)d_in[2];
  const float* lin_W    = (const float*)d_in[3];   // [128, 768]
  const float* sheaf_W  = (const float*)d_in[4];   // [256, 6]
  const float* conv1_W  = (const float*)d_in[5];   // [128, 128]
  const float* conv1_b  = (const float*)d_in[6];   // [128]
  const float* conv2_W  = (const float*)d_in[7];
  const float* conv2_b  = (const float*)d_in[8];
  const float* lin2_W   = (const float*)d_in[9];   // [768, 40]
  float* out = (float*)d_out;

  const int* row = eidx;
  const int* col = eidx + NNZ;

  // --- workspace carve-up (floats), ~150 MB total ---
  float* ws    = (float*)d_ws;
  float* bufA  = ws;                           // 15,360,000  (xl -> later h/out2)
  float* bufB  = bufA + (size_t)NN * DH;       // 15,360,000  (projection buffer)
  float* el    = bufB + (size_t)NN * DH;       //  1,536,000
  float* mbuf  = el + (size_t)EE * DH;         //  1,536,000
  float* xmean = mbuf + (size_t)EE * DH;       //  2,560,000
  float* emean = xmean + (size_t)NN * FF;      //    256,000
  float* xm    = emean + (size_t)EE * FF;      //    120,000
  float* em    = xm + (size_t)NN * DD;         //     12,000
  float* alpha = em + (size_t)EE * DD;         //    600,000
  float* Dg    = alpha + (size_t)NNZ * DD;     //    120,000  (-> Dinv in place)
  float* Bg    = Dg + (size_t)NN * DD;         //     12,000  (-> Binv in place)

  const size_t nBig = (size_t)NN * DH;
  const size_t eBig = (size_t)EE * DH;

  // 1) xl = x @ lin_W  [N, d, F];  el = hedge_attr @ lin_W  [E, d, F]
  launch_gemm(x, lin_W, bufA, NN, DH, FF, DH, DH, stream);
  launch_gemm(hedge, lin_W, el, EE, DH, FF, DH, DH, stream);

  // 2) means over d
  mean_d_kernel<<<(NN * FF + 255) / 256, 256, 0, stream>>>(bufA, xmean, NN);
  mean_d_kernel<<<(EE * FF + 255) / 256, 256, 0, stream>>>(el, emean, EE);

  // 3) xm = xmean @ sheaf_W[:F];  em = emean @ sheaf_W[F:]
  //    (concat([xs,es]) @ sheaf_W decomposed into two K=128 GEMMs)
  launch_gemm(xmean, sheaf_W, xm, NN, DD, FF, DD, DD, stream);
  launch_gemm(emean, sheaf_W + FF * DD, em, EE, DD, FF, DD, DD, stream);

  // 4) alpha + sheaf degrees (static; shared by both conv layers)
  fill_kernel<<<64, 256, 0, stream>>>(Dg, (size_t)NN * DD, 0.0f);
  fill_kernel<<<64, 256, 0, stream>>>(Bg, (size_t)EE * DD, 0.0f);
  alpha_deg_kernel<<<(NNZ * DD + 255) / 256, 256, 0, stream>>>(xm, em, row, col,
                                                               alpha, Dg, Bg);
  recip_kernel<<<(NN * DD + 255) / 256, 256, 0, stream>>>(Dg, NN * DD);
  recip_kernel<<<(EE * DD + 255) / 256, 256, 0, stream>>>(Bg, EE * DD);

  // ---- conv layer 1 ----
  // xp = einsum(xl, conv1_W)  ==  [N*d, F] @ [F, H]
  launch_gemm(bufA, conv1_W, bufB, NN * DD, HH, FF, HH, HH, stream);
  fill_kernel<<<1024, 256, 0, stream>>>(mbuf, eBig, 0.0f);
  scatter_to_edge<<<NNZ, 192, 0, stream>>>(bufB, alpha, Bg, row, col, mbuf);
  fill_kernel<<<4096, 256, 0, stream>>>(bufA, nBig, 0.0f);   // xl dead; reuse as h
  scatter_to_node<<<NNZ, 192, 0, stream>>>(mbuf, alpha, Dg, row, col, bufA);
  bias_act_kernel<<<(unsigned)((nBig + 255) / 256), 256, 0, stream>>>(bufA, conv1_b,
                                                                      nBig, 1);

  // ---- conv layer 2 ----
  launch_gemm(bufA, conv2_W, bufB, NN * DD, HH, HH, HH, HH, stream);
  fill_kernel<<<1024, 256, 0, stream>>>(mbuf, eBig, 0.0f);
  scatter_to_edge<<<NNZ, 192, 0, stream>>>(bufB, alpha, Bg, row, col, mbuf);
  fill_kernel<<<4096, 256, 0, stream>>>(bufA, nBig, 0.0f);   // h dead; reuse as out2
  scatter_to_node<<<NNZ, 192, 0, stream>>>(mbuf, alpha, Dg, row, col, bufA);
  bias_act_kernel<<<(unsigned)((nBig + 255) / 256), 256, 0, stream>>>(bufA, conv2_b,
                                                                      nBig, 0);

  // ---- final: [N, d*H] @ [768, 40] -> d_out ----
  launch_gemm(bufA, lin2_W, out, NN, COUT, DH, COUT, COUT, stream);
}